// MFLayer_16363825397836
// MI455X (gfx1250) — compile-verified
//
#include <hip/hip_runtime.h>

typedef float v2f __attribute__((ext_vector_type(2)));
typedef float v8f __attribute__((ext_vector_type(8)));

#define KDIM 32
#define PDIM 10
#define BDIM 64
#define KP   (KDIM / 2)  // 16 K-pairs
#define UPITCH 80        // float2 per Un2 row (64 used + 16 pad): 160 dwords, %64==32
#define QPITCH 16        // float2 per Qpad2 row: 32 dwords, %64==32

// ---------------------------------------------------------------------------
// Kernel 1: transpose U (64 x N, row-major) -> Ut (N x 64) so each gathered
// neighbor column becomes one contiguous 256B read.
// ---------------------------------------------------------------------------
__global__ __launch_bounds__(256) void mf_transpose_kernel(
    const float* __restrict__ U, float* __restrict__ Ut, int N)
{
    __shared__ float tile[64][65];   // 65 pad: conflict-free transpose
    const int n0 = blockIdx.x * 64;
    const int t  = threadIdx.x;

    #pragma unroll
    for (int i = 0; i < 16; ++i) {
        int idx = t + i * 256;       // 0..4095
        int b   = idx >> 6;          // 0..63
        int nl  = idx & 63;
        int n   = n0 + nl;
        if (n < N) tile[nl][b] = U[(size_t)b * N + n];
    }
    __syncthreads();

    #pragma unroll
    for (int i = 0; i < 16; ++i) {
        int idx = t + i * 256;
        int nl  = idx >> 6;
        int b   = idx & 63;
        int n   = n0 + nl;
        if (n < N) Ut[(size_t)n * BDIM + b] = tile[nl][b];
    }
}

// ---------------------------------------------------------------------------
// Kernel 2: one workgroup (128 threads = 4 waves) per m.
//   LDS is K-pair interleaved so every WMMA operand is ONE ds_load_b64
//   (compiler fuses pairs into ds_load_2addr_b64):
//     Un2[kp][b] = { U[b, ids[2kp]], U[b, ids[2kp+1]] }
//     Qpad2[kp][a] = { Q[m][2kp][a], Q[m][2kp+1][a] }  (a >= P zero-padded)
//   Per-wave WMMA GEMM: (16 x 32) x (32 x 16) in 8 steps of K=4.
//   All global indexing uses 32-bit offsets (tensors < 4 GB) so the backend
//   can use SGPR-base + 32-bit-VGPR-offset addressing.
// ---------------------------------------------------------------------------
__global__ __launch_bounds__(128) void mf_wmma_kernel(
    const float* __restrict__ Usrc,   // Ut (N x 64) if transposed=1, else U (64 x N)
    const float* __restrict__ Q,      // M x K x P
    const int*   __restrict__ ids,    // M x K
    float* __restrict__ out,          // B x M x P
    int N, int M, int transposed)
{
    __shared__ v2f Un2[KP * UPITCH];    // 16 x 80 float2 = 10 KB
    __shared__ v2f Qpad2[KP * QPITCH];  // 16 x 16 float2 = 2 KB

    const int m = blockIdx.x;
    const int t = threadIdx.x;
    const float* Qm  = Q   + (size_t)m * (KDIM * PDIM);  // uniform -> SGPR base
    const int*   idm = ids + (size_t)m * KDIM;           // uniform -> SGPR base

    // Overlap Q fetch with the Un gather: 32 prefetch lanes cover the 320-float row.
    __builtin_prefetch(Qm + (t & 31) * PDIM, 0, 0);

    // ---- stage Qpad2: 256 float2 entries, 2 iterations ----
    #pragma unroll
    for (int i = 0; i < 2; ++i) {
        int pi = t + i * 128;            // 0..255
        int kp = pi >> 4;                // 0..15
        int a  = pi & 15;                // 0..15
        v2f q;
        q[0] = (a < PDIM) ? Qm[(2 * kp    ) * PDIM + a] : 0.0f;
        q[1] = (a < PDIM) ? Qm[(2 * kp + 1) * PDIM + a] : 0.0f;
        Qpad2[kp * QPITCH + a] = q;
    }

    // ---- gather Un2: 16 kp-rows x 64 b = 1024 float2, 8 iterations ----
    #pragma unroll
    for (int i = 0; i < 8; ++i) {
        int j  = t + i * 128;            // 0..1023
        int kp = j >> 6;                 // 0..15
        int b  = j & 63;
        unsigned id0 = (unsigned)idm[2 * kp];
        unsigned id1 = (unsigned)idm[2 * kp + 1];
        v2f v;
        if (transposed) {
            // Ut is N*64 elements (< 2^32): 32-bit offsets, contiguous per kp-row
            v[0] = Usrc[id0 * (unsigned)BDIM + (unsigned)b];
            v[1] = Usrc[id1 * (unsigned)BDIM + (unsigned)b];
        } else {
            // strided fallback straight from U (B x N)
            v[0] = Usrc[(unsigned)b * (unsigned)N + id0];
            v[1] = Usrc[(unsigned)b * (unsigned)N + id1];
        }
        Un2[kp * UPITCH + b] = v;
    }
    __syncthreads();

    // ---- WMMA phase: wave w handles b-rows [16w, 16w+16) ----
    const int w    = t >> 5;
    const int lane = t & 31;
    const int half = lane >> 4;       // 0: K pair {0,1}; 1: K pair {2,3}
    const int lm   = lane & 15;       // A: row M; B: column N
    const int brow = (w << 4) + lm;   // global b for this lane's A rows

    v8f c = {0.f, 0.f, 0.f, 0.f, 0.f, 0.f, 0.f, 0.f};

    #pragma unroll
    for (int kk = 0; kk < KDIM / 4; ++kk) {
        const int kp = kk * 2 + half;    // which K-pair this lane half supplies

        // A fragment (16x4 f32): one ds_load_b64 -> adjacent VGPR pair
        v2f af = Un2[kp * UPITCH + brow];
        // B fragment (4x16 f32): one ds_load_b64 -> adjacent VGPR pair
        v2f bf = Qpad2[kp * QPITCH + lm];

        // 8 args: (neg_a, A, neg_b, B, c_mod, C, reuse_a, reuse_b)
        c = __builtin_amdgcn_wmma_f32_16x16x4_f32(
                false, af, false, bf, (short)0, c, false, false);
    }

    // ---- store phase: D layout: VGPR r -> M = r + 8*half, N = lm ----
    // out is B*M*P = 32M elements (128 MB): 32-bit offsets from uniform base.
    if (lm < PDIM) {
        float* outm = out + (unsigned)m * (unsigned)PDIM;   // uniform -> SGPR base
        const unsigned bstride = (unsigned)M * (unsigned)PDIM;
        unsigned off = (unsigned)((w << 4) + (half << 3)) * bstride + (unsigned)lm;
        #pragma unroll
        for (int r = 0; r < 8; ++r) {
            outm[off] = c[r];
            off += bstride;
        }
    }
}

// ---------------------------------------------------------------------------
extern "C" void kernel_launch(void* const* d_in, const int* in_sizes, int n_in,
                              void* d_out, int out_size, void* d_ws, size_t ws_size,
                              hipStream_t stream) {
    const float* U   = (const float*)d_in[0];   // (B, N) f32
    const float* Q   = (const float*)d_in[1];   // (M, K, P) f32
    const int*   ids = (const int*)d_in[2];     // (M, K) i32
    float*       out = (float*)d_out;           // (B, M, P) f32

    const int B = BDIM, K = KDIM;
    const int N = in_sizes[0] / B;
    const int M = in_sizes[2] / K;

    const size_t ut_bytes = (size_t)N * B * sizeof(float);
    if (ws_size >= ut_bytes) {
        float* Ut = (float*)d_ws;
        const int nblocks = (N + 63) / 64;
        mf_transpose_kernel<<<nblocks, 256, 0, stream>>>(U, Ut, N);
        mf_wmma_kernel<<<M, 128, 0, stream>>>(Ut, Q, ids, out, N, M, 1);
    } else {
        mf_wmma_kernel<<<M, 128, 0, stream>>>(U, Q, ids, out, N, M, 0);
    }
}